// LRMUCell_10831907521069
// MI455X (gfx1250) — compile-verified
//
#include <hip/hip_runtime.h>

// ---------------------------------------------------------------------------
// LRMU cell, restructured:
//   m_T = W @ C,  W[r,t] = (x[b,t,:] @ K)[m]  (r = b*128+m),
//   C[t] = S[255-t],  S[j] = B_vec @ A^j   (built in log depth)
// All GEMMs via V_WMMA_F32_16X16X4_F32 (wave32, 16x16 tile per wave).
// ---------------------------------------------------------------------------

typedef __attribute__((ext_vector_type(2))) float v2f;
typedef __attribute__((ext_vector_type(8))) float v8f;

#define BATCH 16
#define SEQ   256
#define IN_D  128
#define MEM_D 128
#define ORDER 256

// ---------------------------------------------------------------------------
// Stage 1: W[(b*128+m), t] = sum_d K[d,m] * x[b,t,d]
// Per wave: one 16x16 tile of W_b = K^T (128x128) @ x_b^T (128x256), K-loop 32.
// Grid: 16 batches * 8 m-tiles * 16 t-tiles = 2048 waves, 4 waves/block.
// ---------------------------------------------------------------------------
__global__ void lrmu_u_wmma(const float* __restrict__ x,
                            const float* __restrict__ K,
                            float* __restrict__ W) {
    const int wave = threadIdx.x >> 5;
    const int lane = threadIdx.x & 31;
    const int gwave = blockIdx.x * 4 + wave;        // 0..2047 (wave-uniform)
    const int tt = gwave & 15;                      // t-tile
    const int mt = (gwave >> 4) & 7;                // m-tile
    const int b  = gwave >> 7;                      // batch
    const int half = lane >> 4;
    const int lrow = lane & 15;
    const int m0 = mt * 16, t0 = tt * 16;

    v8f acc = {};
    for (int k0 = 0; k0 < IN_D; k0 += 4) {
        const int ka = k0 + 2 * half;
        v2f a, bm;
        // A'[m][k] = K[k][m]  (K row-major IN_D x MEM_D)
        a.x = K[(ka + 0) * MEM_D + (m0 + lrow)];
        a.y = K[(ka + 1) * MEM_D + (m0 + lrow)];
        // B'[k][t] = x[b][t][k]
        const int xrow = (b * SEQ + (t0 + lrow)) * IN_D;
        bm.x = x[xrow + ka + 0];
        bm.y = x[xrow + ka + 1];
        acc = __builtin_amdgcn_wmma_f32_16x16x4_f32(false, a, false, bm,
                                                    (short)0, acc, false, false);
    }
    const int base = (b * MEM_D + m0) * SEQ + t0;
#pragma unroll
    for (int v = 0; v < 8; ++v)
        W[base + (v + 8 * half) * SEQ + lrow] = acc[v];
}

// ---------------------------------------------------------------------------
// Stage 2a (seed): S[0] = B_vec ; S[j] = S[j-1] @ A  for j = 1..15.
// One workgroup, 256 threads, LDS-held previous row. Only 15 serial steps.
// ---------------------------------------------------------------------------
__global__ void lrmu_seed(const float* __restrict__ A,
                          const float* __restrict__ Bvec,
                          float* __restrict__ S) {
    __shared__ float prev[ORDER];
    const int c = threadIdx.x;
    float v = Bvec[c];
    S[c] = v;
    prev[c] = v;
    __syncthreads();
    for (int j = 1; j < 16; ++j) {
        float acc = 0.f;
#pragma unroll 4
        for (int k = 0; k < ORDER; ++k)
            acc = fmaf(prev[k], A[k * ORDER + c], acc);
        __syncthreads();
        prev[c] = acc;
        S[j * ORDER + c] = acc;
        __syncthreads();
    }
}

// ---------------------------------------------------------------------------
// Generic row-major GEMM, all matrices 256 columns wide:
//   O[M x 256] = Am[M x 256] @ Bm[256 x 256],  M = 16*Mtiles, K = 256.
// Used for repeated squaring of A and for doubling the S row-block.
// ---------------------------------------------------------------------------
__global__ void gemm256_wmma(const float* __restrict__ Am,
                             const float* __restrict__ Bm,
                             float* __restrict__ Om, int Mtiles) {
    const int wave = threadIdx.x >> 5;
    const int lane = threadIdx.x & 31;
    const int gwave = blockIdx.x * 4 + wave;        // wave-uniform
    const int nt = gwave & 15;
    const int mt = gwave >> 4;
    if (mt >= Mtiles) return;                       // whole-wave exit: EXEC stays all-1s
    const int half = lane >> 4;
    const int lrow = lane & 15;
    const int m0 = mt * 16, n0 = nt * 16;

    v8f acc = {};
    for (int k0 = 0; k0 < 256; k0 += 4) {
        const int ka = k0 + 2 * half;
        v2f a, bm;
        a.x = Am[(m0 + lrow) * 256 + ka + 0];
        a.y = Am[(m0 + lrow) * 256 + ka + 1];
        bm.x = Bm[(ka + 0) * 256 + n0 + lrow];
        bm.y = Bm[(ka + 1) * 256 + n0 + lrow];
        acc = __builtin_amdgcn_wmma_f32_16x16x4_f32(false, a, false, bm,
                                                    (short)0, acc, false, false);
    }
    const int base = m0 * 256 + n0;
#pragma unroll
    for (int v = 0; v < 8; ++v)
        Om[base + (v + 8 * half) * 256 + lrow] = acc[v];
}

// ---------------------------------------------------------------------------
// Stage 3: out[r,o] = sum_t W[r,t] * S[255-t][o]   (out is 2048 x 256)
// 2048 tiles (128 r-tiles x 16 o-tiles), K-loop 64.
// ---------------------------------------------------------------------------
__global__ void lrmu_final_wmma(const float* __restrict__ W,
                                const float* __restrict__ S,
                                float* __restrict__ out) {
    const int wave = threadIdx.x >> 5;
    const int lane = threadIdx.x & 31;
    const int gwave = blockIdx.x * 4 + wave;        // 0..2047
    const int ot = gwave & 15;
    const int rt = gwave >> 4;
    const int half = lane >> 4;
    const int lrow = lane & 15;
    const int r0 = rt * 16, o0 = ot * 16;

    v8f acc = {};
    for (int k0 = 0; k0 < SEQ; k0 += 4) {
        const int ka = k0 + 2 * half;
        v2f a, bm;
        a.x = W[(r0 + lrow) * SEQ + ka + 0];
        a.y = W[(r0 + lrow) * SEQ + ka + 1];
        // C[t] = S[255 - t]
        bm.x = S[(255 - (ka + 0)) * ORDER + o0 + lrow];
        bm.y = S[(255 - (ka + 1)) * ORDER + o0 + lrow];
        acc = __builtin_amdgcn_wmma_f32_16x16x4_f32(false, a, false, bm,
                                                    (short)0, acc, false, false);
    }
    const int base = r0 * ORDER + o0;
#pragma unroll
    for (int v = 0; v < 8; ++v)
        out[base + (v + 8 * half) * ORDER + lrow] = acc[v];
}

// ---------------------------------------------------------------------------
// Launch: inputs are {x, memory_kernel, A, B_mat} (all f32).
// Workspace: W (2 MB) | S (256 KB) | P0 (256 KB) | P1 (256 KB).
// ---------------------------------------------------------------------------
extern "C" void kernel_launch(void* const* d_in, const int* in_sizes, int n_in,
                              void* d_out, int out_size, void* d_ws, size_t ws_size,
                              hipStream_t stream) {
    (void)in_sizes; (void)n_in; (void)out_size; (void)ws_size;
    const float* x    = (const float*)d_in[0];
    const float* K    = (const float*)d_in[1];
    const float* A    = (const float*)d_in[2];
    const float* Bvec = (const float*)d_in[3];
    float* out = (float*)d_out;

    char* ws = (char*)d_ws;
    float* W  = (float*)(ws);                                  // 2048*256 f32
    float* S  = (float*)(ws + (2u << 20));                     // 256*256 f32
    float* P0 = (float*)(ws + (2u << 20) + (256u << 10));      // 256*256 f32
    float* P1 = (float*)(ws + (2u << 20) + (512u << 10));      // 256*256 f32

    // W = per-batch K^T @ x^T  (independent of the power chain)
    lrmu_u_wmma<<<512, 128, 0, stream>>>(x, K, W);

    // Seed S rows 0..15 (15 short serial steps)
    lrmu_seed<<<1, 256, 0, stream>>>(A, Bvec, S);

    // Repeated squaring: A^2, A^4, A^8, A^16
    gemm256_wmma<<<64, 128, 0, stream>>>(A,  A,  P0, 16);   // A^2
    gemm256_wmma<<<64, 128, 0, stream>>>(P0, P0, P1, 16);   // A^4
    gemm256_wmma<<<64, 128, 0, stream>>>(P1, P1, P0, 16);   // A^8
    gemm256_wmma<<<64, 128, 0, stream>>>(P0, P0, P1, 16);   // A^16

    // Doubling of the S row-block, interleaved with further squarings
    gemm256_wmma<<< 4, 128, 0, stream>>>(S, P1, S +  16 * 256, 1);  // rows 16..31  (@A^16)
    gemm256_wmma<<<64, 128, 0, stream>>>(P1, P1, P0, 16);            // A^32
    gemm256_wmma<<< 8, 128, 0, stream>>>(S, P0, S +  32 * 256, 2);  // rows 32..63  (@A^32)
    gemm256_wmma<<<64, 128, 0, stream>>>(P0, P0, P1, 16);            // A^64
    gemm256_wmma<<<16, 128, 0, stream>>>(S, P1, S +  64 * 256, 4);  // rows 64..127 (@A^64)
    gemm256_wmma<<<64, 128, 0, stream>>>(P1, P1, P0, 16);            // A^128
    gemm256_wmma<<<32, 128, 0, stream>>>(S, P0, S + 128 * 256, 8);  // rows 128..255(@A^128)

    // out = W @ C  with C[t] = S[255-t]
    lrmu_final_wmma<<<512, 128, 0, stream>>>(W, S, out);
}